// VQVAE_17746804867560
// MI455X (gfx1250) — compile-verified
//
#include <hip/hip_runtime.h>
#include <math.h>

#define BN 8192
#define LGB 13
#define EPS 1e-5f

typedef __attribute__((ext_vector_type(2))) float v2f;
typedef __attribute__((ext_vector_type(8))) float v8f;

// Layout: all activation tensors are stored channel-group-of-4 interleaved:
//   float index = ((g*BN + n)*PS + ph*PW + pw)*4 + (c & 3),   g = c >> 2
// with PW = W+4, PS = (H+4)*PW (2-wide zero halo). A position's 4 channels of
// one group are contiguous -> WMMA B-fragments are single b64 loads, D-matrix
// stores are b128, elementwise kernels are float4.

// ---------------------------------------------------------------------------
// Stage input: (8192,1,28,28) -> group 0 plane 36x36, image at [4..31],
// halo and channel slots 1..3 zeroed (float4 {v,0,0,0} everywhere).
// ---------------------------------------------------------------------------
__global__ __launch_bounds__(256) void k_stage_input(const float* __restrict__ x,
                                                     float4* __restrict__ buf) {
  int n = blockIdx.x;
  int ibase = n * 784;
  int obase = n * 1296;
  for (int i = threadIdx.x; i < 1296; i += 256) {
    int ph = i / 36;
    int pw = i - ph * 36;
    float v = 0.f;
    if (ph >= 4 && ph < 32 && pw >= 4 && pw < 32) v = x[ibase + (ph - 4) * 28 + (pw - 4)];
    buf[obase + i] = make_float4(v, 0.f, 0.f, 0.f);
  }
}

// ---------------------------------------------------------------------------
// Zero the 2-wide halo of a grouped tensor. One block per (g,n) plane.
// ---------------------------------------------------------------------------
__global__ __launch_bounds__(256) void k_halo_zero(float4* __restrict__ buf, int H, int W) {
  int PW = W + 4;
  int PS = (H + 4) * PW;
  int base = (int)blockIdx.x * PS;
  float4 z = make_float4(0.f, 0.f, 0.f, 0.f);
  for (int i = threadIdx.x; i < PS; i += 256) {
    int ph = i / PW;
    int pw = i - ph * PW;
    if (ph < 2 || ph >= H + 2 || pw < 2 || pw >= W + 2) buf[base + i] = z;
  }
}

// ---------------------------------------------------------------------------
// 5x5 'same' conv + bias as implicit GEMM on V_WMMA_F32_16X16X4_F32.
//   A (16x4, MxK) = weights (cout x cin-chunk), staged in LDS as fragments
//   B (4x16, KxN) = input patches, one global_load_b64 per tile per chunk
//   C/D (16x16)   = (cout, position) fp32 accumulators, stored as b128
// Each wave owns 32 consecutive linear positions = two tiles sharing each
// A-fragment. Tap offsets are maintained incrementally (no divisions).
// B*H*W is a multiple of 256 -> no tails, EXEC all-ones at every WMMA.
// ---------------------------------------------------------------------------
template <int CIN, int COUT>
__global__ __launch_bounds__(256) void k_conv5x5(const float* __restrict__ src,
                                                 float* __restrict__ dst,
                                                 const float* __restrict__ Wt,
                                                 const float* __restrict__ bias,
                                                 int lgW, int lgHW) {
  constexpr int NCK = (CIN + 3) / 4;      // K-chunks (channel groups) per tap
  constexpr int NCHUNK = 25 * NCK;
  __shared__ float lw[NCHUNK * 64];       // A-fragments: [chunk][lane][2]

  const int Wd = 1 << lgW;
  const int HW = 1 << lgHW;
  const int Hd = HW >> lgW;
  const int PW = Wd + 4;
  const int PS = (Hd + 4) * PW;
  const int GS = BN * PS * 4;             // group stride in floats

  // A-fragment layout: lanes 0-15: M=lane, VGPR0=K0, VGPR1=K1;
  //                    lanes16-31: M=lane-16, VGPR0=K2, VGPR1=K3
  for (int idx = threadIdx.x; idx < NCHUNK * 64; idx += 256) {
    int chunk = idx >> 6;
    int slot = idx & 63;
    int lane = slot >> 1;
    int j = slot & 1;
    int tap = chunk / NCK;
    int ck = chunk - tap * NCK;
    int m = lane & 15;
    int ci = ck * 4 + ((lane < 16) ? 0 : 2) + j;
    float v = 0.f;
    if (m < COUT && ci < CIN) v = Wt[(m * CIN + ci) * 25 + tap];
    lw[chunk * 64 + lane * 2 + j] = v;
  }
  __syncthreads();

  const int lane = threadIdx.x & 31;
  const int waveId = blockIdx.x * 8 + (threadIdx.x >> 5);
  const int nl = lane & 15;
  const int p0 = waveId * 32 + nl;
  const int p1 = p0 + 16;

  int img0 = p0 >> lgHW, rem0 = p0 & (HW - 1);
  int pb0 = img0 * PS + ((rem0 >> lgW) + 2) * PW + ((rem0 & (Wd - 1)) + 2);
  int img1 = p1 >> lgHW, rem1 = p1 & (HW - 1);
  int pb1 = img1 * PS + ((rem1 >> lgW) + 2) * PW + ((rem1 & (Wd - 1)) + 2);

  const int kSel = (lane < 16) ? 0 : 2;   // which K-pair this lane supplies
  const int pq0 = pb0 * 4 + kSel;         // float index of B pair, tile 0
  const int pq1 = pb1 * 4 + kSel;

  v8f acc0 = {}, acc1 = {};
  int off = (-2 * PW - 2) * 4;            // tap offset in floats
  int chunk = 0;
  for (int dh = 0; dh < 5; ++dh) {
#pragma unroll
    for (int dw = 0; dw < 5; ++dw) {
#pragma unroll
      for (int ck = 0; ck < NCK; ++ck) {
        v2f a = *(const v2f*)&lw[chunk * 64 + lane * 2];
        v2f b0 = *(const v2f*)(src + ck * GS + pq0 + off);
        v2f b1 = *(const v2f*)(src + ck * GS + pq1 + off);
        acc0 = __builtin_amdgcn_wmma_f32_16x16x4_f32(false, a, false, b0, (short)0, acc0,
                                                     false, false);
        acc1 = __builtin_amdgcn_wmma_f32_16x16x4_f32(false, a, false, b1, (short)0, acc1,
                                                     false, false);
        ++chunk;
      }
      off += 4;
    }
    off += (PW - 5) * 4;
  }

  // D layout: VGPR r -> M = r (lanes 0-15) / r+8 (lanes 16-31), N = lane&15.
  // Rows 4t..4t+3 of this lane-half = channel group gBase+t -> float4 store.
  const int gBase = (lane < 16) ? 0 : 2;
#pragma unroll
  for (int t = 0; t < 2; ++t) {
    int g = gBase + t;
    if (COUT > g * 4) {
      float4 v0, v1;
      float b0v = (g * 4 + 0 < COUT) ? bias[g * 4 + 0] : 0.f;
      float b1v = (g * 4 + 1 < COUT) ? bias[g * 4 + 1] : 0.f;
      float b2v = (g * 4 + 2 < COUT) ? bias[g * 4 + 2] : 0.f;
      float b3v = (g * 4 + 3 < COUT) ? bias[g * 4 + 3] : 0.f;
      v0.x = (g * 4 + 0 < COUT) ? acc0[4 * t + 0] + b0v : 0.f;
      v0.y = (g * 4 + 1 < COUT) ? acc0[4 * t + 1] + b1v : 0.f;
      v0.z = (g * 4 + 2 < COUT) ? acc0[4 * t + 2] + b2v : 0.f;
      v0.w = (g * 4 + 3 < COUT) ? acc0[4 * t + 3] + b3v : 0.f;
      v1.x = (g * 4 + 0 < COUT) ? acc1[4 * t + 0] + b0v : 0.f;
      v1.y = (g * 4 + 1 < COUT) ? acc1[4 * t + 1] + b1v : 0.f;
      v1.z = (g * 4 + 2 < COUT) ? acc1[4 * t + 2] + b2v : 0.f;
      v1.w = (g * 4 + 3 < COUT) ? acc1[4 * t + 3] + b3v : 0.f;
      *(float4*)(dst + g * GS + pb0 * 4) = v0;
      *(float4*)(dst + g * GS + pb1 * 4) = v1;
    }
  }
}

// ---------------------------------------------------------------------------
// BatchNorm batch statistics, deterministic two-stage tree reduction.
// ---------------------------------------------------------------------------
__global__ __launch_bounds__(256) void k_stats_partial(const float* __restrict__ src,
                                                       float2* __restrict__ part,
                                                       int lgW, int lgHW) {
  const int c = blockIdx.y;             // real channel
  const int g = c >> 2, cl = c & 3;
  const int Wd = 1 << lgW;
  const int HW = 1 << lgHW;
  const int Hd = HW >> lgW;
  const int PW = Wd + 4;
  const int PS = (Hd + 4) * PW;
  const int N = BN << lgHW;
  const int stride = gridDim.x * 256;
  float s = 0.f, sq = 0.f;
  for (int p = blockIdx.x * 256 + threadIdx.x; p < N; p += stride) {
    int img = p >> lgHW;
    int rem = p & (HW - 1);
    int h = rem >> lgW;
    int w = rem & (Wd - 1);
    float v = src[((g * BN + img) * PS + (h + 2) * PW + (w + 2)) * 4 + cl];
    s += v;
    sq += v * v;
  }
  __shared__ float ls[256], lq[256];
  ls[threadIdx.x] = s;
  lq[threadIdx.x] = sq;
  __syncthreads();
  for (int o = 128; o > 0; o >>= 1) {
    if (threadIdx.x < o) {
      ls[threadIdx.x] += ls[threadIdx.x + o];
      lq[threadIdx.x] += lq[threadIdx.x + o];
    }
    __syncthreads();
  }
  if (threadIdx.x == 0) part[c * gridDim.x + blockIdx.x] = make_float2(ls[0], lq[0]);
}

__global__ __launch_bounds__(256) void k_stats_final(const float2* __restrict__ part,
                                                     float2* __restrict__ ss,
                                                     const float* __restrict__ g,
                                                     const float* __restrict__ be,
                                                     int NB, int N) {
  const int c = blockIdx.x;
  float s = 0.f, sq = 0.f;
  for (int i = threadIdx.x; i < NB; i += 256) {
    float2 v = part[c * NB + i];
    s += v.x;
    sq += v.y;
  }
  __shared__ float ls[256], lq[256];
  ls[threadIdx.x] = s;
  lq[threadIdx.x] = sq;
  __syncthreads();
  for (int o = 128; o > 0; o >>= 1) {
    if (threadIdx.x < o) {
      ls[threadIdx.x] += ls[threadIdx.x + o];
      lq[threadIdx.x] += lq[threadIdx.x + o];
    }
    __syncthreads();
  }
  if (threadIdx.x == 0) {
    float inv_n = 1.f / (float)N;
    float mean = ls[0] * inv_n;
    float var = lq[0] * inv_n - mean * mean;
    var = fmaxf(var, 0.f);
    float scale = g[c] * rsqrtf(var + EPS);
    ss[c] = make_float2(scale, be[c] - mean * scale);
  }
}

// One thread = one float4 (4 channels of a group at one position), in place.
__global__ __launch_bounds__(256) void k_bn_tanh(float4* __restrict__ buf,
                                                 const float2* __restrict__ ss,
                                                 int lgW, int lgHW) {
  int idx = blockIdx.x * 256 + threadIdx.x;   // over NG*BN*HW
  int lgBHW = lgHW + LGB;
  int gI = idx >> lgBHW;
  int p = idx & ((1 << lgBHW) - 1);
  int Wd = 1 << lgW;
  int HW = 1 << lgHW;
  int Hd = HW >> lgW;
  int PW = Wd + 4;
  int PS = (Hd + 4) * PW;
  int img = p >> lgHW;
  int rem = p & (HW - 1);
  int h = rem >> lgW;
  int w = rem & (Wd - 1);
  int a = (gI * BN + img) * PS + (h + 2) * PW + (w + 2);
  float2 s0 = ss[gI * 4 + 0], s1 = ss[gI * 4 + 1], s2 = ss[gI * 4 + 2], s3 = ss[gI * 4 + 3];
  float4 v = buf[a];
  v.x = tanhf(v.x * s0.x + s0.y);
  v.y = tanhf(v.y * s1.x + s1.y);
  v.z = tanhf(v.z * s2.x + s2.y);
  v.w = tanhf(v.w * s3.x + s3.y);
  buf[a] = v;
}

// ---------------------------------------------------------------------------
// 2x2 max-pool and nearest x2 upsample (grouped float4 layout).
// ---------------------------------------------------------------------------
__global__ __launch_bounds__(256) void k_maxpool2(const float4* __restrict__ src,
                                                  float4* __restrict__ dst,
                                                  int lgW2, int lgHW2) {
  int idx = blockIdx.x * 256 + threadIdx.x;   // over NG*BN*H2*W2
  int lgBHW = lgHW2 + LGB;
  int gI = idx >> lgBHW;
  int p = idx & ((1 << lgBHW) - 1);
  int W2 = 1 << lgW2;
  int HW2 = 1 << lgHW2;
  int H2 = HW2 >> lgW2;
  int img = p >> lgHW2;
  int rem = p & (HW2 - 1);
  int h2 = rem >> lgW2;
  int w2 = rem & (W2 - 1);
  int PWd = W2 + 4, PSd = (H2 + 4) * PWd;
  int Ws = W2 * 2, Hs = H2 * 2;
  int PWs = Ws + 4, PSs = (Hs + 4) * PWs;
  int sb = (gI * BN + img) * PSs + (2 * h2 + 2) * PWs + (2 * w2 + 2);
  float4 a = src[sb], b = src[sb + 1], c = src[sb + PWs], d = src[sb + PWs + 1];
  float4 m;
  m.x = fmaxf(fmaxf(a.x, b.x), fmaxf(c.x, d.x));
  m.y = fmaxf(fmaxf(a.y, b.y), fmaxf(c.y, d.y));
  m.z = fmaxf(fmaxf(a.z, b.z), fmaxf(c.z, d.z));
  m.w = fmaxf(fmaxf(a.w, b.w), fmaxf(c.w, d.w));
  dst[(gI * BN + img) * PSd + (h2 + 2) * PWd + (w2 + 2)] = m;
}

__global__ __launch_bounds__(256) void k_upsample2(const float4* __restrict__ src,
                                                   float4* __restrict__ dst,
                                                   int lgW, int lgHW) {
  int idx = blockIdx.x * 256 + threadIdx.x;   // over NG*BN*H*W (dst dims)
  int lgBHW = lgHW + LGB;
  int gI = idx >> lgBHW;
  int p = idx & ((1 << lgBHW) - 1);
  int Wd = 1 << lgW;
  int HW = 1 << lgHW;
  int Hd = HW >> lgW;
  int img = p >> lgHW;
  int rem = p & (HW - 1);
  int h = rem >> lgW;
  int w = rem & (Wd - 1);
  int PW = Wd + 4, PS = (Hd + 4) * PW;
  int Ws = Wd >> 1, Hs = Hd >> 1;
  int PWs = Ws + 4, PSs = (Hs + 4) * PWs;
  dst[(gI * BN + img) * PS + (h + 2) * PW + (w + 2)] =
      src[(gI * BN + img) * PSs + ((h >> 1) + 2) * PWs + ((w >> 1) + 2)];
}

// ---------------------------------------------------------------------------
// Vector quantization. Faithful to z.view(-1,16) on NCHW: row (n*16+c) is one
// channel's 4x4 spatial map. Writes zq into decoder-input buffer + loss parts.
// ---------------------------------------------------------------------------
__global__ __launch_bounds__(256) void k_vq(const float* __restrict__ z,
                                            float* __restrict__ zq,
                                            const float* __restrict__ cb,
                                            float* __restrict__ lossPart) {
  __shared__ float scb[80];  // 5 x 16
  if (threadIdx.x < 80) scb[threadIdx.x] = cb[threadIdx.x];
  __syncthreads();

  int idx = blockIdx.x * 256 + threadIdx.x;  // n*16 + c, total 8192*16
  int n = idx >> 4;
  int c = idx & 15;
  int g = c >> 2, cl = c & 3;
  const int PW = 8, PS = 64;
  int base = ((g * BN + n) * PS + 2 * PW + 2) * 4 + cl;

  float zv[16];
#pragma unroll
  for (int j = 0; j < 16; ++j) zv[j] = z[base + ((j >> 2) * PW + (j & 3)) * 4];

  float best = 3.402823466e38f;
  int bi = 0;
#pragma unroll
  for (int e = 0; e < 5; ++e) {
    float d = 0.f;
#pragma unroll
    for (int j = 0; j < 16; ++j) {
      float t = zv[j] - scb[e * 16 + j];
      d += t * t;
    }
    if (d < best) {
      best = d;
      bi = e;
    }
  }

  float lsum = 0.f;
#pragma unroll
  for (int j = 0; j < 16; ++j) {
    float q = scb[bi * 16 + j];
    zq[base + ((j >> 2) * PW + (j & 3)) * 4] = q;
    float t = q - zv[j];
    lsum += t * t;
  }

  __shared__ float red[256];
  red[threadIdx.x] = lsum;
  __syncthreads();
  for (int o = 128; o > 0; o >>= 1) {
    if (threadIdx.x < o) red[threadIdx.x] += red[threadIdx.x + o];
    __syncthreads();
  }
  if (threadIdx.x == 0) lossPart[blockIdx.x] = red[0];
}

__global__ __launch_bounds__(512) void k_loss_final(const float* __restrict__ part, int nb,
                                                    float* __restrict__ out) {
  __shared__ float red[512];
  float s = 0.f;
  for (int i = threadIdx.x; i < nb; i += 512) s += part[i];
  red[threadIdx.x] = s;
  __syncthreads();
  for (int o = 256; o > 0; o >>= 1) {
    if (threadIdx.x < o) red[threadIdx.x] += red[threadIdx.x + o];
    __syncthreads();
  }
  // loss = codebook_loss + commitment_loss = 2 * mean((zq - z)^2)
  if (threadIdx.x == 0) out[0] = 2.f * red[0] / (float)(BN * 16 * 16);
}

// ---------------------------------------------------------------------------
// Final: 1ch 16x16 post-tanh activation -> nearest x2 upsample -> crop 28x28
// ---------------------------------------------------------------------------
__global__ __launch_bounds__(256) void k_final_out(const float* __restrict__ act,
                                                   float* __restrict__ out) {
  int idx = blockIdx.x * 256 + threadIdx.x;  // 8192*784
  int n = idx / 784;
  int rem = idx - n * 784;
  int h = rem / 28;
  int w = rem - h * 28;
  int hs = (h + 2) >> 1;
  int ws = (w + 2) >> 1;
  out[idx] = act[(n * 400 + (hs + 2) * 20 + (ws + 2)) * 4];  // plane 20x20, cl=0
}

// ---------------------------------------------------------------------------
// Host orchestration
// ---------------------------------------------------------------------------
static void launch_conv(int Cin, int Cout, int blocks, hipStream_t s, const float* src,
                        float* dst, const float* Wt, const float* bias, int lgW, int lgHW) {
#define CCASE(CI, CO)                                                            \
  if (Cin == CI && Cout == CO) {                                                 \
    k_conv5x5<CI, CO><<<blocks, 256, 0, s>>>(src, dst, Wt, bias, lgW, lgHW);     \
    return;                                                                      \
  }
  CCASE(1, 4) CCASE(4, 4) CCASE(4, 8) CCASE(8, 8) CCASE(8, 16) CCASE(16, 16)
  CCASE(16, 8) CCASE(8, 4) CCASE(4, 1)
#undef CCASE
}

extern "C" void kernel_launch(void* const* d_in, const int* in_sizes, int n_in,
                              void* d_out, int out_size, void* d_ws, size_t ws_size,
                              hipStream_t stream) {
  (void)in_sizes; (void)n_in; (void)out_size; (void)ws_size;

  const float* x = (const float*)d_in[0];
  // enc_params: 16 layers x {W,b,g,be} at d_in[1..64]; dec_params at d_in[65..112]
  const float* cb = (const float*)d_in[113];

  static const int ENC_CI[16] = {1, 4, 4, 4, 4, 8, 8, 8, 8, 16, 16, 16, 16, 16, 16, 16};
  static const int ENC_CO[16] = {4, 4, 4, 4, 8, 8, 8, 8, 16, 16, 16, 16, 16, 16, 16, 16};
  static const int DEC_CI[12] = {16, 16, 16, 16, 8, 8, 8, 8, 4, 4, 4, 4};
  static const int DEC_CO[12] = {16, 16, 16, 8, 8, 8, 8, 4, 4, 4, 4, 1};

  float* wsf = (float*)d_ws;
  const size_t BUFSZ = (size_t)1 * BN * 36 * 36 * 4;  // 42,467,328 floats (largest: 1 group @36x36)
  float* bufX = wsf;
  float* bufY = wsf + BUFSZ;
  float* partials = wsf + 2 * BUFSZ;            // 16*256*2
  float* scaleshift = partials + 16 * 256 * 2;  // 16 float2
  float* vqpart = scaleshift + 32;              // 512

  float* cur = bufX;
  float* nxt = bufY;
  auto flip = [&]() { float* t = cur; cur = nxt; nxt = t; };

  const int NB = 256;

  auto conv_layer = [&](int Cin, int Cout, int H, int W, const float* Wt, const float* bs,
                        const float* g, const float* be) {
    int lgW = __builtin_ctz(W);
    int lgHW = __builtin_ctz(H * W);
    int NG = (Cout + 3) / 4;
    k_halo_zero<<<NG * BN, 256, 0, stream>>>((float4*)nxt, H, W);
    int blocks = (BN * H * W) / 256;  // 8 waves/block * 32 positions/wave
    launch_conv(Cin, Cout, blocks, stream, cur, nxt, Wt, bs, lgW, lgHW);
    k_stats_partial<<<dim3(NB, Cout), 256, 0, stream>>>(nxt, (float2*)partials, lgW, lgHW);
    k_stats_final<<<Cout, 256, 0, stream>>>((const float2*)partials, (float2*)scaleshift, g,
                                            be, NB, BN << lgHW);
    int total = NG * BN * H * W;      // float4s
    k_bn_tanh<<<total / 256, 256, 0, stream>>>((float4*)nxt, (const float2*)scaleshift, lgW,
                                               lgHW);
    flip();
  };

  // ---- encoder ----
  k_stage_input<<<BN, 256, 0, stream>>>(x, (float4*)cur);
  int H = 32, W = 32;
  for (int i = 0; i < 16; ++i) {
    const float* Wt = (const float*)d_in[1 + 4 * i + 0];
    const float* bs = (const float*)d_in[1 + 4 * i + 1];
    const float* g  = (const float*)d_in[1 + 4 * i + 2];
    const float* be = (const float*)d_in[1 + 4 * i + 3];
    conv_layer(ENC_CI[i], ENC_CO[i], H, W, Wt, bs, g, be);
    if (i == 3 || i == 7 || i == 11) {
      int NG = ENC_CO[i] / 4;
      int H2 = H / 2, W2 = W / 2;
      k_halo_zero<<<NG * BN, 256, 0, stream>>>((float4*)nxt, H2, W2);
      int total = NG * BN * H2 * W2;
      k_maxpool2<<<total / 256, 256, 0, stream>>>((const float4*)cur, (float4*)nxt,
                                                  __builtin_ctz(W2), __builtin_ctz(H2 * W2));
      flip();
      H = H2;
      W = W2;
    }
  }
  // cur = z: 16ch (4 groups) @ 4x4

  // ---- vector quantization ----
  k_halo_zero<<<4 * BN, 256, 0, stream>>>((float4*)nxt, 4, 4);
  k_vq<<<(BN * 16) / 256, 256, 0, stream>>>(cur, nxt, cb, vqpart);
  k_loss_final<<<1, 512, 0, stream>>>(vqpart, (BN * 16) / 256,
                                      (float*)d_out + (size_t)BN * 784);
  flip();  // cur = zq

  // ---- decoder ----
  H = 4;
  W = 4;
  for (int j = 0; j < 12; ++j) {
    const float* Wt = (const float*)d_in[65 + 4 * j + 0];
    const float* bs = (const float*)d_in[65 + 4 * j + 1];
    const float* g  = (const float*)d_in[65 + 4 * j + 2];
    const float* be = (const float*)d_in[65 + 4 * j + 3];
    conv_layer(DEC_CI[j], DEC_CO[j], H, W, Wt, bs, g, be);
    if (j == 3 || j == 7) {
      int NG = DEC_CO[j] / 4;
      int H2 = H * 2, W2 = W * 2;
      k_halo_zero<<<NG * BN, 256, 0, stream>>>((float4*)nxt, H2, W2);
      int total = NG * BN * H2 * W2;
      k_upsample2<<<total / 256, 256, 0, stream>>>((const float4*)cur, (float4*)nxt,
                                                   __builtin_ctz(W2), __builtin_ctz(H2 * W2));
      flip();
      H = H2;
      W = W2;
    }
    // j == 11: final upsample x2 + crop fused into k_final_out
  }
  // cur = 1ch (group 0) @ 16x16 post-tanh; upsample x2 + center crop fused:
  k_final_out<<<(BN * 784) / 256, 256, 0, stream>>>(cur, (float*)d_out);
}